// MNN_65755949302191
// MI455X (gfx1250) — compile-verified
//
#include <hip/hip_runtime.h>
#include <hip/hip_bf16.h>

// ---------------------------------------------------------------------------
// Problem constants (from the reference)
// ---------------------------------------------------------------------------
#define IHW   306
#define NPIX  (IHW * IHW)          // 93,636 pixels
#define FH    300                  // valid conv output size (306-7+1)
#define KK    7
#define NTAP  (KK * KK)            // 49
#define NCH   10                   // out channels
#define WSROW 32                   // padded channel row in workspace (20 used)
#define NB1   368                  // kernel-1 blocks; multiple of 4 for WMMA chunks
#define POOLN (NCH * FH * 30)      // 90,000   pooled elements
#define MAPN  (NCH * FH * FH)      // 900,000  per broadcast map
#define OUTSZ (POOLN + 2 * MAPN)   // 1,890,000 output floats

typedef __attribute__((ext_vector_type(2))) float v2f;
typedef __attribute__((ext_vector_type(8))) float v8f;

// ---------------------------------------------------------------------------
// Kernel 1: per-pixel clipped accumulation + deterministic block reduction.
// Each thread owns one pixel x[a,b]. For each of the 49 taps (u,v) the pixel
// contributes to output position (a-u, b-v) iff that position is in [0,300)².
// Branchless: invalid taps replace x with +1e30 (kills min(.,0)) / -1e30
// (kills max(.,0)). All K_hit/K_miss indices are wave-uniform -> s_load.
// ---------------------------------------------------------------------------
__global__ __launch_bounds__(256)
void mnn_partial_kernel(const float* __restrict__ x,
                        const float* __restrict__ khit,
                        const float* __restrict__ kmiss,
                        float* __restrict__ ws)
{
    const int tid = threadIdx.x;
    const int pid = blockIdx.x * 256 + tid;
    const int a   = pid / IHW;           // pid >= NPIX -> a >= 306 -> all taps invalid
    const int b   = pid - a * IHW;
    const float xv = (pid < NPIX) ? x[pid] : 0.0f;

    float accH[NCH], accM[NCH];
#pragma unroll
    for (int o = 0; o < NCH; ++o) { accH[o] = 0.0f; accM[o] = 0.0f; }

    for (int u = 0; u < KK; ++u) {
        const bool rok = (a >= u) && (a <= (FH - 1) + u);
        for (int v = 0; v < KK; ++v) {
            const bool ok = rok && (b >= v) && (b <= (FH - 1) + v);
            const float xh = ok ? xv :  1.0e30f;   // min(xh - k, 0) == 0 when invalid
            const float xm = ok ? xv : -1.0e30f;   // max(xm - k, 0) == 0 when invalid
            const int t = u * KK + v;
#pragma unroll
            for (int o = 0; o < NCH; ++o) {
                accH[o] += fminf(xh - khit [o * NTAP + t], 0.0f);
                accM[o] += fmaxf(xm - kmiss[o * NTAP + t], 0.0f);
            }
        }
    }

    // Deterministic block reduction: 20 channels x 256 lanes in LDS
    // (row stride 257 words -> conflict-free serial sums), threads 0..19 each
    // sum one channel in fixed order; threads 20..31 write zero padding.
    __shared__ float red[20][257];
#pragma unroll
    for (int o = 0; o < NCH; ++o) {
        red[o     ][tid] = accH[o];
        red[o + 10][tid] = accM[o];
    }
    __syncthreads();
    if (tid < WSROW) {
        float s = 0.0f;
        if (tid < 20) {
            for (int i = 0; i < 256; ++i) s += red[tid][i];
        }
        ws[(size_t)blockIdx.x * WSROW + tid] = s;   // channels 20..31 -> 0 pad
    }
}

// ---------------------------------------------------------------------------
// Kernel 2: reduce the 368x32 partials matrix to 20 channel sums with chained
// V_WMMA_F32_16X16X4_F32:  D = A(16x4) * ones(4x16) + D  ->  per-row sums.
// Half 0 covers channels 0..15, half 1 covers channels 16..31 (20..31 are the
// zero padding). Single wave32, uniform control flow (EXEC all ones at WMMA).
// A layout (32-bit A 16x4): lanes 0-15 hold K=0,1 in the two VGPRs for row
// M=lane; lanes 16-31 hold K=2,3 for row M=lane-16.
// ---------------------------------------------------------------------------
__global__ __launch_bounds__(32)
void mnn_wmma_reduce_kernel(const float* __restrict__ ws,
                            float* __restrict__ vals)
{
    const int L    = threadIdx.x;
    const int m    = L & 15;
    const int koff = (L < 16) ? 0 : 2;

    v2f bones; bones.x = 1.0f; bones.y = 1.0f;
    v8f d0 = {};   // channels  0..15
    v8f d1 = {};   // channels 16..31 (16..19 real, rest zero)

    for (int i = 0; i < NB1; i += 4) {
        const float* base = ws + (size_t)(i + koff) * WSROW;
        v2f a0, a1;
        a0.x = base[m];                 // partial row i+koff   , channel m
        a0.y = base[WSROW + m];         // partial row i+koff+1 , channel m
        a1.x = base[16 + m];            // channel 16+m
        a1.y = base[WSROW + 16 + m];
        d0 = __builtin_amdgcn_wmma_f32_16x16x4_f32(false, a0, false, bones,
                                                   (short)0, d0, false, false);
        d1 = __builtin_amdgcn_wmma_f32_16x16x4_f32(false, a1, false, bones,
                                                   (short)0, d1, false, false);
    }

    // D layout: (VGPR r, lane L) holds M = r + (L>=16 ? 8 : 0), N = L%16.
    // Every N column carries the same row sum; lanes with N==0 export rows.
    __shared__ float sums[32];
    if (m == 0) {
        const int mbase = (L < 16) ? 0 : 8;
#pragma unroll
        for (int r = 0; r < 8; ++r) sums[mbase + r]      = d0[r];
#pragma unroll
        for (int r = 0; r < 8; ++r) sums[16 + mbase + r] = d1[r];
    }
    __syncthreads();
    if (L < NCH) {
        const float hit  = sums[L];        // channels  0..9  = F_hit_sum
        const float miss = sums[10 + L];   // channels 10..19 = F_miss_sum
        vals[L]           = hit;
        vals[10 + L]      = miss;
        vals[20 + L]      = hit - miss;    // F_map value (pool of a constant)
    }
}

// ---------------------------------------------------------------------------
// Kernel 3: broadcast-fill the three outputs with b128 stores (7.56 MB total,
// pure store-bandwidth). Layout: pooled(10,300,30) | hit(10,300,300) |
// miss(10,300,300). All segment/channel sizes divisible by 4.
// ---------------------------------------------------------------------------
__global__ __launch_bounds__(256)
void mnn_fill_kernel(const float* __restrict__ vals, float* __restrict__ out)
{
    const int t = blockIdx.x * 256 + threadIdx.x;
    const int e = t * 4;
    if (e >= OUTSZ) return;
    float v;
    if (e < POOLN) {
        v = vals[20 + e / (FH * 30)];                    // F_map per channel
    } else if (e < POOLN + MAPN) {
        v = vals[(e - POOLN) / (FH * FH)];               // F_hit_sum
    } else {
        v = vals[10 + (e - (POOLN + MAPN)) / (FH * FH)]; // F_miss_sum
    }
    float4 q = make_float4(v, v, v, v);
    *reinterpret_cast<float4*>(out + e) = q;             // global_store_b128
}

// ---------------------------------------------------------------------------
// Launch: partials -> WMMA reduction -> broadcast fill (in-order on stream).
// Workspace: NB1*WSROW partial floats + 30 result floats ≈ 47 KB, fully
// rewritten every call (deterministic, no reliance on prior state).
// ---------------------------------------------------------------------------
extern "C" void kernel_launch(void* const* d_in, const int* in_sizes, int n_in,
                              void* d_out, int out_size, void* d_ws, size_t ws_size,
                              hipStream_t stream)
{
    const float* x     = (const float*)d_in[0];   // (1,1,306,306)
    const float* khit  = (const float*)d_in[1];   // (10,1,7,7)
    const float* kmiss = (const float*)d_in[2];   // (10,1,7,7)
    float* ws   = (float*)d_ws;
    float* vals = ws + (size_t)NB1 * WSROW;
    float* out  = (float*)d_out;

    mnn_partial_kernel<<<NB1, 256, 0, stream>>>(x, khit, kmiss, ws);
    mnn_wmma_reduce_kernel<<<1, 32, 0, stream>>>(ws, vals);

    const int fillThreads = OUTSZ / 4;                       // 472,500
    const int fillBlocks  = (fillThreads + 255) / 256;       // 1,846
    mnn_fill_kernel<<<fillBlocks, 256, 0, stream>>>(vals, out);
}